// DroplessFusedGating_50019189129736
// MI455X (gfx1250) — compile-verified
//
#include <hip/hip_runtime.h>

// ---------------------------------------------------------------------------
// Dropless fused MoE gating for MI455X (gfx1250, wave32).
// Outputs (concatenated in d_out): weight_out [S,6] f32 | loss [1] f32 |
// topk_idx [S,6] i32 (stored as raw int32 bits).
// d_ws: gate_sum[160] f32 | hist[160] i32.
// ---------------------------------------------------------------------------

#define NUM_E   160
#define TOPK    6
#define NEG_INF (-__builtin_inff())

typedef __attribute__((ext_vector_type(2))) float v2f;
typedef __attribute__((ext_vector_type(4))) float v4f;
typedef __attribute__((ext_vector_type(8))) float v8f;

// ---------------------------------------------------------------------------
// Zero the workspace accumulators (must run every call: graph replay safe).
// ---------------------------------------------------------------------------
__global__ void moe_zero_ws(float* __restrict__ gsum, int* __restrict__ hist) {
    int t = threadIdx.x;
    if (t < NUM_E) { gsum[t] = 0.0f; hist[t] = 0; }
}

// ---------------------------------------------------------------------------
// Main gating kernel: one wave32 per token (grid-stride over tokens).
// Lane l owns experts {4l,4l+1,4l+2,4l+3, 128+l}.
// ---------------------------------------------------------------------------
__global__ __launch_bounds__(256) void moe_gating(
        const float* __restrict__ x,
        float* __restrict__ wout,          // [S, 6]
        int*   __restrict__ iout,          // [S, 6]
        float* __restrict__ g_gsum,        // [160]
        int*   __restrict__ g_hist,        // [160]
        int S) {
    __shared__ float s_gsum[NUM_E];
    __shared__ int   s_hist[NUM_E];

    const int tid = threadIdx.x;
    if (tid < NUM_E) { s_gsum[tid] = 0.0f; s_hist[tid] = 0; }
    __syncthreads();

    const int lane    = tid & 31;
    const int wavesPB = blockDim.x >> 5;
    const int wid     = blockIdx.x * wavesPB + (tid >> 5);
    const int nwaves  = gridDim.x * wavesPB;

    // Per-wave gate_sum accumulators (one per owned expert slot).
    float acc0 = 0.f, acc1 = 0.f, acc2 = 0.f, acc3 = 0.f, acc4 = 0.f;

    for (int token = wid; token < S; token += nwaves) {
        const float* row = x + (size_t)token * NUM_E;

        // Prefetch next row this wave will process (global_prefetch_b8).
        if (token + nwaves < S)
            __builtin_prefetch(x + (size_t)(token + nwaves) * NUM_E + lane * 4, 0, 0);

        // Coalesced load: global_load_b128 + global_load_b32 per token.
        v4f  q  = *(const v4f*)(row + lane * 4);
        float g0 = q.x, g1 = q.y, g2 = q.z, g3 = q.w;
        float g4 = row[128 + lane];

        // --- softmax: wave max ---
        float m = fmaxf(fmaxf(fmaxf(g0, g1), fmaxf(g2, g3)), g4);
        #pragma unroll
        for (int off = 16; off > 0; off >>= 1)
            m = fmaxf(m, __shfl_xor(m, off, 32));

        g0 = __expf(g0 - m); g1 = __expf(g1 - m); g2 = __expf(g2 - m);
        g3 = __expf(g3 - m); g4 = __expf(g4 - m);

        // --- softmax: wave sum ---
        float s = g0 + g1 + g2 + g3 + g4;
        #pragma unroll
        for (int off = 16; off > 0; off >>= 1)
            s += __shfl_xor(s, off, 32);
        float inv = 1.0f / s;
        g0 *= inv; g1 *= inv; g2 *= inv; g3 *= inv; g4 *= inv;

        acc0 += g0; acc1 += g1; acc2 += g2; acc3 += g3; acc4 += g4;

        // --- top-6 via 6 masked wave argmax passes ---
        float t0 = g0, t1 = g1, t2 = g2, t3 = g3, t4 = g4;
        float wv = 0.0f; int wi = 0;
        #pragma unroll
        for (int t = 0; t < TOPK; ++t) {
            // local argmax over 5 owned experts (strict > keeps lowest index)
            float bv = t0; int be = 4 * lane;
            if (t1 > bv) { bv = t1; be = 4 * lane + 1; }
            if (t2 > bv) { bv = t2; be = 4 * lane + 2; }
            if (t3 > bv) { bv = t3; be = 4 * lane + 3; }
            if (t4 > bv) { bv = t4; be = 128 + lane;   }
            // wave argmax allreduce, tie-break to lower expert index
            #pragma unroll
            for (int off = 16; off > 0; off >>= 1) {
                float ov = __shfl_xor(bv, off, 32);
                int   oi = __shfl_xor(be, off, 32);
                if (ov > bv || (ov == bv && oi < be)) { bv = ov; be = oi; }
            }
            if (lane == t) { wv = bv; wi = be; }
            // mask the winner in its owner lane
            if (be < 128) {
                if (lane == (be >> 2)) {
                    int sl = be & 3;
                    if      (sl == 0) t0 = NEG_INF;
                    else if (sl == 1) t1 = NEG_INF;
                    else if (sl == 2) t2 = NEG_INF;
                    else              t3 = NEG_INF;
                }
            } else if (lane == (be - 128)) {
                t4 = NEG_INF;
            }
        }

        // lanes 0..5 write this token's outputs + LDS histogram bump
        if (lane < TOPK) {
            size_t o = (size_t)token * TOPK + lane;
            wout[o] = wv * 16.0f;            // ROUTED_SCALING_FACTOR
            iout[o] = wi;
            atomicAdd(&s_hist[wi], 1);       // ds_add_u32
        }
    }

    // flush per-wave gate_sum accumulators into block LDS (ds_add_f32)
    atomicAdd(&s_gsum[4 * lane + 0], acc0);
    atomicAdd(&s_gsum[4 * lane + 1], acc1);
    atomicAdd(&s_gsum[4 * lane + 2], acc2);
    atomicAdd(&s_gsum[4 * lane + 3], acc3);
    atomicAdd(&s_gsum[128 + lane],   acc4);
    __syncthreads();

    // one global atomic per expert per block
    if (tid < NUM_E) {
        atomicAdd(&g_gsum[tid], s_gsum[tid]);
        atomicAdd(&g_hist[tid], s_hist[tid]);
    }
}

// ---------------------------------------------------------------------------
// Loss epilogue: loss = cap * dot(hist, gate_sum) over 160 experts,
// computed with V_WMMA_F32_16X16X4_F32 (40 accumulating K=4 steps).
// hist chunk broadcast across all A rows; gsum chunk broadcast across all
// B columns -> every element of D equals the dot product.
// One wave, EXEC all-ones (required for WMMA).
// ---------------------------------------------------------------------------
__global__ __launch_bounds__(32) void moe_loss(
        const float* __restrict__ gsum,
        const int*   __restrict__ hist,
        float* __restrict__ loss_out,
        float cap) {
    const int lane = threadIdx.x;
    const int half = (lane >> 4) & 1;       // 0: K=0,1 ; 1: K=2,3 (ISA 16x4 A layout)

    v8f c = {};
    #pragma unroll
    for (int ch = 0; ch < NUM_E / 4; ++ch) {
        const int base = ch * 4 + 2 * half;
        v2f a, b;
        a.x = (float)hist[base];
        a.y = (float)hist[base + 1];
        b.x = gsum[base];
        b.y = gsum[base + 1];
        // D = A x B + C  (16x16x4 f32)
        c = __builtin_amdgcn_wmma_f32_16x16x4_f32(
                /*neg_a=*/false, a, /*neg_b=*/false, b,
                /*c_mod=*/(short)0, c, /*reuse_a=*/false, /*reuse_b=*/false);
    }
    if (lane == 0) loss_out[0] = cap * c[0];
}

// ---------------------------------------------------------------------------
extern "C" void kernel_launch(void* const* d_in, const int* in_sizes, int n_in,
                              void* d_out, int out_size, void* d_ws, size_t ws_size,
                              hipStream_t stream) {
    (void)n_in; (void)out_size; (void)ws_size;

    const float* x = (const float*)d_in[0];
    const int S = in_sizes[0] / NUM_E;     // 131072

    float* wout = (float*)d_out;                              // [S,6]
    float* loss = wout + (size_t)S * TOPK;                    // [1]
    int*   iout = ((int*)d_out) + (size_t)S * TOPK + 1;       // [S,6] int32 bits

    float* gsum = (float*)d_ws;                               // [160]
    int*   hist = (int*)(gsum + NUM_E);                       // [160]

    moe_zero_ws<<<1, 256, 0, stream>>>(gsum, hist);

    const int blocks = 2048;   // 16384 waves -> 8 grid-stride tokens per wave
    moe_gating<<<blocks, 256, 0, stream>>>(x, wout, iout, gsum, hist, S);

    const float cap = (float)((double)NUM_E /
                              ((double)S * (double)S * (double)TOPK));
    moe_loss<<<1, 32, 0, stream>>>(gsum, hist, loss, cap);
}